// canny_edge_27101243637980
// MI455X (gfx1250) — compile-verified
//
#include <hip/hip_runtime.h>
#include <stdint.h>

// MI455X (gfx1250) bandwidth-bound fused Canny pipeline.
// CDNA5 paths: global_load_async_to_lds_b32 staging (ASYNCcnt) and
// v_wmma_f32_16x16x4_f32 chains for the separable 5x5 Gaussian blur.

typedef __attribute__((ext_vector_type(2))) float v2f;
typedef __attribute__((ext_vector_type(8))) float v8f;

#define HSZ 512
#define WSZ 512
#define PLANE (HSZ * WSZ)   // 262144

__device__ __forceinline__ int reflect_i(int i) {
  i = (i < 0) ? -i : i;
  i = (i > HSZ - 1) ? (2 * (HSZ - 1) - i) : i;
  return i;
}
__device__ __forceinline__ int clamp_i(int i) {
  return (i < 0) ? 0 : ((i > HSZ - 1) ? (HSZ - 1) : i);
}

// CDNA5 async global->LDS gather (per-lane LDS dest + 64-bit global addr).
// Low 32 bits of a generic LDS pointer are the LDS byte offset (aperture in [63:32]).
__device__ __forceinline__ void async_ld(const float* g, const float* lds) {
  unsigned loff = (unsigned)(uintptr_t)lds;
  unsigned long long ga = (unsigned long long)(uintptr_t)g;
  asm volatile("global_load_async_to_lds_b32 %0, %1, off"
               :: "v"(loff), "v"(ga) : "memory");
}
__device__ __forceinline__ void wait_async0() {
  asm volatile("s_wait_asynccnt 0" ::: "memory");
}

// 5-tap Gaussian (ksize=5, sigma=1), normalized.
__device__ __forceinline__ float gcoef(int d) {
  float v = 0.f;
  if (d == 0 || d == 4) v = 0.054488685f;
  if (d == 1 || d == 3) v = 0.244201342f;
  if (d == 2)           v = 0.402619947f;
  return v;
}
// Banded weight: W[k][n] = g[k-n] for 0<=k-n<=4 else 0
__device__ __forceinline__ float band(int k, int n) {
  int d = k - n;
  return (d >= 0 && d <= 4) ? gcoef(d) : 0.f;
}

// ---------------- gray = 0.299*x[2] + 0.587*x[1] + 0.114*x[0] --------------
__global__ __launch_bounds__(256) void k_gray(const float* __restrict__ x,
                                              float* __restrict__ gray) {
  int i = blockIdx.x * 256 + threadIdx.x;    // float4 index, 32*65536 total
  int b = i >> 16;
  int p = i & 65535;
  const float4* x4 = (const float4*)x;
  float4 r  = x4[((size_t)b * 3 + 2) * 65536 + p];
  float4 g  = x4[((size_t)b * 3 + 1) * 65536 + p];
  float4 bl = x4[((size_t)b * 3 + 0) * 65536 + p];
  float4 o;
  o.x = 0.299f * r.x + 0.587f * g.x + 0.114f * bl.x;
  o.y = 0.299f * r.y + 0.587f * g.y + 0.114f * bl.y;
  o.z = 0.299f * r.z + 0.587f * g.z + 0.114f * bl.z;
  o.w = 0.299f * r.w + 0.587f * g.w + 0.114f * bl.w;
  ((float4*)gray)[i] = o;
}

// ---------------- 5x5 Gaussian blur via WMMA f32 16x16x4 -------------------
// Per 16x16 tile: H = E(16x20) x W(20x16)  (two 16-row chunks),
// then out = W'(16x20) x Hmat(20x16). Each matmul = 5 chained K=4 WMMAs.
__global__ __launch_bounds__(256) void k_blur(const float* __restrict__ gray,
                                              float* __restrict__ blur) {
  __shared__ float sE[8][32 * 21];   // rows rel [-4,28), cols rel [-2,18), stride 21
  __shared__ float sH[8][32 * 17];   // H rows rel [-4,28), 16 cols, stride 17
  const int tid  = threadIdx.x;
  const int lane = tid & 31;
  const int wv   = tid >> 5;
  const int tile = blockIdx.x * 8 + wv;         // 32768 tiles total
  const int tc = tile & 31, tr = (tile >> 5) & 31, b = tile >> 10;
  const int r0 = tr * 16, c0 = tc * 16;
  const float* gb = gray + (size_t)b * PLANE;

  float* eb = &sE[wv][0];
  float* hb = &sH[wv][0];

  // async-stage E with reflect padding (blur uses 'reflect' on gray)
  for (int e = lane; e < 32 * 20; e += 32) {    // uniform 20 iterations
    int r = e / 20, c = e - r * 20;
    int gr = reflect_i(r0 + r - 4);
    int gc = reflect_i(c0 + c - 2);
    async_ld(gb + gr * WSZ + gc, eb + r * 21 + c);
  }
  wait_async0();

  const int half = lane >> 4;       // A: K split; B: K split; C/D: row half
  const int m    = lane & 15;       // A row / B col / C-D col

  // Banded coefficient chunks; identical per-lane values for horizontal-B
  // (W[k][n]=g[k-n], n=lane&15) and vertical-A (W'[m][k]=g[k-m], m=lane&15).
  v2f wc[5];
#pragma unroll
  for (int j = 0; j < 5; ++j) {
    int kb = 4 * j + half * 2;
    wc[j].x = band(kb, m);
    wc[j].y = band(kb + 1, m);
  }

  // Horizontal pass: two 16-row chunks -> H in LDS
#pragma unroll
  for (int q = 0; q < 2; ++q) {
    v8f acc = {0.f, 0.f, 0.f, 0.f, 0.f, 0.f, 0.f, 0.f};
#pragma unroll
    for (int j = 0; j < 5; ++j) {
      int kb = 4 * j + half * 2;
      v2f a;
      a.x = eb[(q * 16 + m) * 21 + kb];
      a.y = eb[(q * 16 + m) * 21 + kb + 1];
      acc = __builtin_amdgcn_wmma_f32_16x16x4_f32(false, a, false, wc[j],
                                                  (short)0, acc, false, false);
    }
#pragma unroll
    for (int r = 0; r < 8; ++r)
      hb[(q * 16 + half * 8 + r) * 17 + m] = acc[r];
  }

  // Vertical pass: out = W' x Hmat, Hmat[k] = H rel row (k-2) = hb row (k+2)
  v8f acc = {0.f, 0.f, 0.f, 0.f, 0.f, 0.f, 0.f, 0.f};
#pragma unroll
  for (int j = 0; j < 5; ++j) {
    int kb = 4 * j + half * 2;
    v2f bm;
    bm.x = hb[(kb + 2) * 17 + m];
    bm.y = hb[(kb + 3) * 17 + m];
    acc = __builtin_amdgcn_wmma_f32_16x16x4_f32(false, wc[j], false, bm,
                                                (short)0, acc, false, false);
  }
  float* ob = blur + (size_t)b * PLANE;
#pragma unroll
  for (int r = 0; r < 8; ++r)
    ob[(size_t)(r0 + half * 8 + r) * WSZ + (c0 + m)] = acc[r];
}

// ---------------- Sobel + magnitude + NMS ----------------------------------
__global__ __launch_bounds__(256) void k_mag(const float* __restrict__ blur,
                                             float* __restrict__ mag_out) {
  __shared__ float sB[36 * 37];   // blurred rel [-2,34)^2, edge-clamped, stride 37
  __shared__ float sM[34 * 35];   // mag rel [-1,33)^2 (0 outside image), stride 35
  const int tid = threadIdx.x;
  const int t = blockIdx.x;
  const int tc = t & 15, tr = (t >> 4) & 15, b = t >> 8;
  const int r0 = tr * 32, c0 = tc * 32;
  const float* gb = blur + (size_t)b * PLANE;

  for (int e = tid; e < 36 * 36; e += 256) {
    int r = e / 36, c = e - r * 36;
    int gr = clamp_i(r0 + r - 2), gc = clamp_i(c0 + c - 2);
    async_ld(gb + gr * WSZ + gc, &sB[r * 37 + c]);
  }
  wait_async0();
  __syncthreads();

  for (int e = tid; e < 34 * 34; e += 256) {
    int r = e / 34, c = e - r * 34;
    int gr = r0 + r - 1, gc = c0 + c - 1;
    float mg = 0.f;
    if (gr >= 0 && gr < HSZ && gc >= 0 && gc < WSZ) {
      const float* p = &sB[(r + 1) * 37 + (c + 1)];
      float bmm = p[-38], bm0 = p[-37], bmp = p[-36];
      float b0m = p[-1],  b0p = p[1];
      float bpm = p[36],  bp0 = p[37],  bpp = p[38];
      float gx = (bmp - bmm) + 2.f * (b0p - b0m) + (bpp - bpm);
      float gy = (bpm - bmm) + 2.f * (bp0 - bm0) + (bpp - bmp);
      mg = sqrtf(gx * gx + gy * gy + 1e-6f);
    }
    sM[r * 35 + c] = mg;
  }
  __syncthreads();

  for (int e = tid; e < 32 * 32; e += 256) {
    int r = e >> 5, c = e & 31;
    const float* p = &sB[(r + 2) * 37 + (c + 2)];
    float bmm = p[-38], bm0 = p[-37], bmp = p[-36];
    float b0m = p[-1],  b0p = p[1];
    float bpm = p[36],  bp0 = p[37],  bpp = p[38];
    float gx = (bmp - bmm) + 2.f * (b0p - b0m) + (bpp - bpm);
    float gy = (bpm - bmm) + 2.f * (bp0 - bm0) + (bpp - bmp);
    // idx = round(atan2/(pi/4)) mod 8 ; rintf = round-half-even like jnp.round
    int k = (int)rintf(atan2f(gy, gx) * 1.2732395447351628f) & 7;
    // packed (d+1) 2-bit LUTs: dx={1,1,0,-1,-1,-1,0,1}, dy={0,-1,-1,-1,0,1,1,1}
    int dx = ((0x901A >> (2 * k)) & 3) - 1;
    int dy = ((0xA901 >> (2 * k)) & 3) - 1;
    float mg = sM[(r + 1) * 35 + (c + 1)];
    float np = sM[(r + 1 + dy) * 35 + (c + 1 + dx)];
    float nn = sM[(r + 1 - dy) * 35 + (c + 1 - dx)];
    float o = (fminf(mg - np, mg - nn) > 0.f) ? mg : 0.f;
    mag_out[(size_t)b * PLANE + (size_t)(r0 + r) * WSZ + (c0 + c)] = o;
  }
}

// ---------------- fused separable dilations 3/5/7/9 ------------------------
__global__ __launch_bounds__(256) void k_dilate(const float* __restrict__ mag,
                                                float* __restrict__ o3,
                                                float* __restrict__ o5,
                                                float* __restrict__ o7,
                                                float* __restrict__ o9) {
  __shared__ float sIn[40 * 41];     // mag rel [-4,36)^2, 0 outside, stride 41
  __shared__ float sR[4][40 * 32];   // row-max for half-widths 1..4
  const int tid = threadIdx.x;
  const int t = blockIdx.x;
  const int tc = t & 15, tr = (t >> 4) & 15, b = t >> 8;
  const int r0 = tr * 32, c0 = tc * 32;
  const float* gb = mag + (size_t)b * PLANE;

  for (int e = tid; e < 40 * 40; e += 256) {
    int r = e / 40, c = e - r * 40;
    int gr = r0 + r - 4, gc = c0 + c - 4;
    if (gr >= 0 && gr < HSZ && gc >= 0 && gc < WSZ)
      async_ld(gb + gr * WSZ + gc, &sIn[r * 41 + c]);
    else
      sIn[r * 41 + c] = 0.f;   // -inf pad == 0 since mag >= 0
  }
  wait_async0();
  __syncthreads();

  for (int e = tid; e < 40 * 32; e += 256) {
    int r = e >> 5, c = e & 31;
    const float* p = &sIn[r * 41 + c + 4];
    float v1 = fmaxf(fmaxf(p[-1], p[0]), p[1]);
    float v2 = fmaxf(v1, fmaxf(p[-2], p[2]));
    float v3 = fmaxf(v2, fmaxf(p[-3], p[3]));
    float v4 = fmaxf(v3, fmaxf(p[-4], p[4]));
    sR[0][e] = v1; sR[1][e] = v2; sR[2][e] = v3; sR[3][e] = v4;
  }
  __syncthreads();

  for (int e = tid; e < 32 * 32; e += 256) {
    int r = e >> 5, c = e & 31;
    size_t oi = (size_t)b * PLANE + (size_t)(r0 + r) * WSZ + (c0 + c);
    const float* q0 = &sR[0][(r + 4) * 32 + c];
    const float* q1 = &sR[1][(r + 4) * 32 + c];
    const float* q2 = &sR[2][(r + 4) * 32 + c];
    const float* q3 = &sR[3][(r + 4) * 32 + c];
    float a3 = fmaxf(fmaxf(q0[-32], q0[0]), q0[32]);
    float a5 = fmaxf(fmaxf(fmaxf(q1[-64], q1[-32]), fmaxf(q1[0], q1[32])), q1[64]);
    float a7 = a5;  // independent accumulation below (keep exact per-array maxes)
    a7 = fmaxf(fmaxf(q2[-96], q2[-64]), fmaxf(q2[-32], q2[0]));
    a7 = fmaxf(a7, fmaxf(fmaxf(q2[32], q2[64]), q2[96]));
    float a9 = fmaxf(fmaxf(q3[-128], q3[-96]), fmaxf(q3[-64], q3[-32]));
    a9 = fmaxf(a9, fmaxf(fmaxf(q3[0], q3[32]), fmaxf(q3[64], q3[96])));
    a9 = fmaxf(a9, q3[128]);
    o3[oi] = a3; o5[oi] = a5; o7[oi] = a7; o9[oi] = a9;
  }
}

// ---------------- channel permute (2,0,1) ----------------------------------
__global__ __launch_bounds__(256) void k_perm(const float* __restrict__ x,
                                              float* __restrict__ out5) {
  int i = blockIdx.x * 256 + threadIdx.x;   // float4 index, 96*65536 total
  int p = i & 65535;
  int cb = i >> 16;           // b*3 + c_out
  int b = cb / 3, c_out = cb - b * 3;
  int c_in = (c_out == 0) ? 2 : (c_out - 1);
  ((float4*)out5)[i] = ((const float4*)x)[(size_t)(b * 3 + c_in) * 65536 + p];
}

extern "C" void kernel_launch(void* const* d_in, const int* in_sizes, int n_in,
                              void* d_out, int out_size, void* d_ws, size_t ws_size,
                              hipStream_t stream) {
  (void)in_sizes; (void)n_in; (void)out_size; (void)d_ws; (void)ws_size;
  const float* x = (const float*)d_in[0];
  float* out = (float*)d_out;
  float* mag  = out;                 // 8388608 each
  float* o3   = out + 8388608;
  float* o5   = out + 16777216;
  float* o7   = out + 25165824;
  float* o9   = out + 33554432;
  float* out5 = out + 41943040;      // 25165824 floats
  // scratch lives inside out5 region; k_perm (last) overwrites it
  float* gray = out5;
  float* blur = out5 + 8388608;

  k_gray  <<<8192, 256, 0, stream>>>(x, gray);
  k_blur  <<<4096, 256, 0, stream>>>(gray, blur);
  k_mag   <<<8192, 256, 0, stream>>>(blur, mag);
  k_dilate<<<8192, 256, 0, stream>>>(mag, o3, o5, o7, o9);
  k_perm  <<<24576, 256, 0, stream>>>(x, out5);
}